// CompositeRenderer_11948599018140
// MI455X (gfx1250) — compile-verified
//
#include <hip/hip_runtime.h>
#include <math.h>

// MI455X / gfx1250, wave32. Bandwidth-bound elementwise BRDF:
//   ~168 MB total traffic @ 23.3 TB/s -> ~7.2 us floor; compute ~2 us -> memory-bound.
// Strategy: 4 elements/thread, all streams as b128; normal+viewdir staged through
// LDS with GLOBAL_LOAD_ASYNC_TO_LDS_B128 (ASYNCcnt) + s_wait_asynccnt; prefetch
// ahead into GL2. Clearcoat input is dead in the reference -> never loaded.
// This round: raw v_sqrt_f32 / v_rcp_f32 via builtins to shrink the VALU stream.

#define TPB 256

__device__ __forceinline__ float pow5f(float x) { float x2 = x * x; return x2 * x2 * x; }
__device__ __forceinline__ float satf(float x) { return fminf(fmaxf(x, 0.0f), 1.0f); }
__device__ __forceinline__ float fsqrt(float x) { return __builtin_amdgcn_sqrtf(x); }
__device__ __forceinline__ float frcp(float x)  { return __builtin_amdgcn_rcpf(x); }

__global__ __launch_bounds__(TPB) void brdf_stream_kernel(
    const float* __restrict__ light_p,
    const float* __restrict__ dist_p,
    const float* __restrict__ nrm_p,
    const float* __restrict__ vdir_p,
    const float* __restrict__ anis_p,
    const float* __restrict__ rough_p,
    const float* __restrict__ met_p,
    const float* __restrict__ tint_p,
    const float* __restrict__ salb_p,
    const float* __restrict__ dalb_p,
    float* __restrict__ out_p,
    int nquads)  // N/4 (N divisible by 4)
{
    __shared__ float4 lds_n[TPB * 3];
    __shared__ float4 lds_v[TPB * 3];

    const int t = threadIdx.x;
    const long long q = (long long)blockIdx.x * TPB + t;  // quad index (4 elems)
    if (q >= nquads) return;

    const float4* n4 = ((const float4*)nrm_p) + 3 * q;
    const float4* v4 = ((const float4*)vdir_p) + 3 * q;

    // LDS byte offset = low 32 bits of the generic pointer (aperture lives in hi32).
    unsigned ln = (unsigned)(uintptr_t)(&lds_n[3 * t]);
    unsigned lv = (unsigned)(uintptr_t)(&lds_v[3 * t]);
    unsigned long long gn = (unsigned long long)(uintptr_t)n4;
    unsigned long long gv = (unsigned long long)(uintptr_t)v4;

    // Async global->LDS (INST_OFFSET applies to both LDS and memory address).
    asm volatile("global_load_async_to_lds_b128 %0, %1, off"            :: "v"(ln), "v"(gn) : "memory");
    asm volatile("global_load_async_to_lds_b128 %0, %1, off offset:16"  :: "v"(ln), "v"(gn) : "memory");
    asm volatile("global_load_async_to_lds_b128 %0, %1, off offset:32"  :: "v"(ln), "v"(gn) : "memory");
    asm volatile("global_load_async_to_lds_b128 %0, %1, off"            :: "v"(lv), "v"(gv) : "memory");
    asm volatile("global_load_async_to_lds_b128 %0, %1, off offset:16"  :: "v"(lv), "v"(gv) : "memory");
    asm volatile("global_load_async_to_lds_b128 %0, %1, off offset:32"  :: "v"(lv), "v"(gv) : "memory");

    // Direct-to-VGPR b128 streams (issued while async engine fills LDS).
    float4 d4 = ((const float4*)dist_p)[q];
    float4 a4 = ((const float4*)anis_p)[q];
    float4 r4 = ((const float4*)rough_p)[q];
    float4 m4 = ((const float4*)met_p)[q];
    float4 t4 = ((const float4*)tint_p)[q];
    float4 sa[3], da[3];
    {
        const float4* s4 = ((const float4*)salb_p) + 3 * q;
        const float4* e4 = ((const float4*)dalb_p) + 3 * q;
        sa[0] = s4[0]; sa[1] = s4[1]; sa[2] = s4[2];
        da[0] = e4[0]; da[1] = e4[1]; da[2] = e4[2];
    }
    float L = light_p[0];                 // uniform -> scalar load
    float rcpL = frcp(L);

    // Warm GL2 a few tiles ahead (speculative prefetch; dropped if OOB/unmapped).
    if ((long long)blockIdx.x + 8 < (long long)gridDim.x) {
        __builtin_prefetch(n4 + 3ll * 8 * TPB, 0, 0);
        __builtin_prefetch(v4 + 3ll * 8 * TPB, 0, 0);
    }

    asm volatile("s_wait_asynccnt 0" ::: "memory");

    float4 nq_[3] = { lds_n[3 * t + 0], lds_n[3 * t + 1], lds_n[3 * t + 2] };
    float4 vq_[3] = { lds_v[3 * t + 0], lds_v[3 * t + 1], lds_v[3 * t + 2] };

    const float* nf  = (const float*)nq_;
    const float* vf  = (const float*)vq_;
    const float* saf = (const float*)sa;
    const float* daf = (const float*)da;
    const float* df  = (const float*)&d4;
    const float* af  = (const float*)&a4;
    const float* rf  = (const float*)&r4;
    const float* mf  = (const float*)&m4;
    const float* tf  = (const float*)&t4;

    float4 o_[3];
    float* of = (float*)o_;

    const float ETA   = 1.5f;
    const float PI_   = 3.14159265358979323846f;
    const float INVPI = 0.31830988618379067154f;

#pragma unroll
    for (int k = 0; k < 4; ++k) {
        float anis  = fmaxf(af[k], 1e-5f);
        float rough = fmaxf(rf[k], 1e-5f);
        float met   = fmaxf(mf[k], 1e-5f);
        float tint  = fmaxf(tf[k], 1e-6f);
        float dist  = df[k];
        float nx = nf[3 * k + 0], ny = nf[3 * k + 1], nz = nf[3 * k + 2];
        float vx = vf[3 * k + 0], vy = vf[3 * k + 1], vz = vf[3 * k + 2];
        float s0 = fmaxf(saf[3 * k + 0], 1e-5f);
        float s1 = fmaxf(saf[3 * k + 1], 1e-5f);
        float s2 = fmaxf(saf[3 * k + 2], 1e-5f);
        float b0 = fmaxf(daf[3 * k + 0], 1e-5f);
        float b1 = fmaxf(daf[3 * k + 1], 1e-5f);
        float b2 = fmaxf(daf[3 * k + 2], 1e-5f);

        float c  = nx * vx + ny * vy + nz * vz;
        float c2 = c * c;
        float ac = fabsf(c);

        // anisotropic GGX alphas
        float r2      = fsqrt(rough);
        float aspect  = fsqrt(fmaxf(1.0f - 0.9f * anis, 0.0f));
        float alpha_u = fmaxf(r2 * frcp(aspect), 1e-4f);
        float alpha_v = fmaxf(r2 * aspect, 1e-4f);

        // fresnel_dielectric(c, 1.5)
        float scl = (c > 0.0f) ? (1.0f / ETA) : ETA;
        float ctq = 1.0f - (1.0f - c2) * scl * scl;
        float ctf = fsqrt(fmaxf(ctq, 0.0f));
        float rs  = (ac - ETA * ctf) * frcp(ac + ETA * ctf);
        float rp  = (ETA * ac - ctf) * frcp(ETA * ac + ctf);
        float f_die = 0.5f * (rs * rs + rp * rp);

        // D with alpha = eta (faithful to reference quirk); 1/(2.25+1e-10) folded.
        float a2     = ETA * ETA;
        float root   = c2 + (1.0f - c2) * (1.0f / (a2 + 1e-10f));
        float d_spec = frcp(PI_ * a2 * root * root + 1e-10f);

        // G = g1(c,au)*g1(c,av), denominators fused into one rcp
        float sin_t = fsqrt(fmaxf(1.0f - c2, 0.0f));
        float tan_t = sin_t * frcp(c + 1e-10f);
        float ru = alpha_u * tan_t, rv = alpha_v * tan_t;
        float g_spec = 4.0f * frcp(
            (1.0f + fsqrt(ru * ru + 1.0f)) * (1.0f + fsqrt(rv * rv + 1.0f)));

        // 1/intensity = (d^2+eps)/light  (lum > 0 always)
        float invl = (dist * dist + 1e-10f) * rcpL;

        // calc_schlick: eta clamps -> e = rcp = 0.99999, eta_it < 1 always ->
        // schlick(r0) = wct + r0*(1-wct), wct shared by both calls.
        const float kk = 0.99999f;
        float cts = fsqrt(fmaxf(1.0f - (1.0f - c2) * (kk * kk), 0.0f));
        float wct = pow5f(satf(1.0f - cts));
        float omw = 1.0f - wct;

        float omm   = 1.0f - met;
        float f0c   = 0.04f * invl;                 // ((eta_it-1)/(eta_it+1))^2 == 0.04 both ways
        float basec = omm * (1.0f - tint) * f_die;
        float sc    = 0.25f * d_spec * g_spec * frcp(ac);

        // diffuse + retro
        float fw = pow5f(satf(1.0f - ac));
        float fd = 1.0f - 0.5f * fw; fd *= fd;
        float ad = 1.0f - rough;
        float rr = 2.0f * ad * c2;
        float fr = rr * (2.0f * fw + fw * fw * (rr - 1.0f));
        float dc = omm * ac * INVPI * (fd + fr);

        float fs0 = (wct + s0 * omw) * met + omm * tint * (wct + (f0c * s0) * omw);
        float fs1 = (wct + s1 * omw) * met + omm * tint * (wct + (f0c * s1) * omw);
        float fs2 = (wct + s2 * omw) * met + omm * tint * (wct + (f0c * s2) * omw);

        of[3 * k + 0] = dc * b0 + (basec * s0 + fs0) * sc;
        of[3 * k + 1] = dc * b1 + (basec * s1 + fs1) * sc;
        of[3 * k + 2] = dc * b2 + (basec * s2 + fs2) * sc;
    }

    float4* o4 = ((float4*)out_p) + 3 * q;
    o4[0] = o_[0]; o4[1] = o_[1]; o4[2] = o_[2];
}

extern "C" void kernel_launch(void* const* d_in, const int* in_sizes, int n_in,
                              void* d_out, int out_size, void* d_ws, size_t ws_size,
                              hipStream_t stream) {
    (void)n_in; (void)out_size; (void)d_ws; (void)ws_size;
    const float* light = (const float*)d_in[0];
    const float* dist  = (const float*)d_in[1];
    const float* nrm   = (const float*)d_in[2];
    const float* vdir  = (const float*)d_in[3];
    const float* anis  = (const float*)d_in[4];
    const float* rough = (const float*)d_in[5];
    const float* met   = (const float*)d_in[6];
    // d_in[7] = clearcoat: dead in the reference (branch disabled) -> not loaded
    const float* tint  = (const float*)d_in[8];
    const float* salb  = (const float*)d_in[9];
    const float* dalb  = (const float*)d_in[10];
    float* out = (float*)d_out;

    const int n = in_sizes[1];          // N (distance has N elements)
    const int nquads = n >> 2;          // N divisible by 4 (N = 2^21)
    const int blocks = (nquads + TPB - 1) / TPB;

    hipLaunchKernelGGL(brdf_stream_kernel, dim3(blocks), dim3(TPB), 0, stream,
                       light, dist, nrm, vdir, anis, rough, met, tint,
                       salb, dalb, out, nquads);
}